// Sensate_76905684402762
// MI455X (gfx1250) — compile-verified
//
#include <hip/hip_runtime.h>
#include <math.h>

// ---------------- problem constants ----------------
#define VOCAB 50000
#define S_ 5
#define D_ 256
#define B_ 1024
#define T_ 11
#define SIGMA_ 3.0f
#define LS_ 0.1f
#define A_W2V_ 1.0f
#define A_ORTH_ 0.15f
#define A_ENT_ 0.02f
#define A_L2_ 1e-4f

// GEMM tiling: 64 row-tiles of 16 (B=1024), V split exactly: 50000/16 = 3125 col tiles.
#define NVS 25            // V slices (grid.x)
#define CT_PER_SLICE 125  // 3125 / 25
#define WAVES_PER_BLK 4
#define ITERS 32          // ceil(125/4), uniform loop count for all waves
#define NB2 1024          // blocks for L2 reduction

// ---------------- workspace layout (floats) ----------------
// pooled [B*D] | w2v [B] | ent [B] | orth [B] | l2blk [NB2] | part [NVS*B*4]
#define OFF_POOLED 0
#define OFF_W2V    (B_ * D_)
#define OFF_ENT    (OFF_W2V + B_)
#define OFF_ORTH   (OFF_ENT + B_)
#define OFF_L2B    (OFF_ORTH + B_)
#define OFF_PART   (OFF_L2B + NB2)

typedef float v8f __attribute__((ext_vector_type(8)));
typedef float v2f __attribute__((ext_vector_type(2)));

// deterministic block-wide sum (fixed tree order)
__device__ inline float blockReduceSum(float v, float* red) {
  int tid = threadIdx.x;
  red[tid] = v;
  __syncthreads();
  for (int s = 128; s > 0; s >>= 1) {
    if (tid < s) red[tid] += red[tid + s];
    __syncthreads();
  }
  float r = red[0];
  __syncthreads();
  return r;
}

// ================= kernel 1: gating / pooled / ent / orth =================
// one block (256 threads == D) per batch row
__global__ __launch_bounds__(256) void gating_kernel(
    const float* __restrict__ SE, const int* __restrict__ center_pos,
    const int* __restrict__ qids, float* __restrict__ pooled_out,
    float* __restrict__ ent_out, float* __restrict__ orth_out) {
  __shared__ float red[256];
  __shared__ int ids_s[T_ - 1];
  __shared__ float w_s[T_ - 1];
  int b = blockIdx.x, d = threadIdx.x;
  int cp = center_pos[b];
  int centerId = qids[b * T_ + cp];
  if (d < T_ - 1) {
    int pos = d + ((d >= cp) ? 1 : 0);
    ids_s[d] = qids[b * T_ + pos];
    float dist = (float)(pos - cp);
    w_s[d] = __expf(-(dist * dist) / (2.f * SIGMA_ * SIGMA_));
  }
  __syncthreads();

  float wsum = 0.f;
  for (int t = 0; t < T_ - 1; ++t) wsum += w_s[t];
  float inv_wsum = 1.f / wsum;

  // summary[d] = sum_t w[t] * mean_s SE[ctx_t, s, d]
  float summary = 0.f;
  for (int t = 0; t < T_ - 1; ++t) {
    size_t base = (size_t)ids_s[t] * (S_ * D_) + d;
    float acc = 0.f;
    for (int s = 0; s < S_; ++s) acc += SE[base + (size_t)s * D_];
    summary += w_s[t] * inv_wsum * (acc * (1.f / (float)S_));
  }

  float cs[S_];
  size_t cbase = (size_t)centerId * (S_ * D_) + d;
  for (int s = 0; s < S_; ++s) cs[s] = SE[cbase + (size_t)s * D_];

  float scores[S_];
  for (int s = 0; s < S_; ++s)
    scores[s] = blockReduceSum(cs[s] * summary, red) * (1.f / 16.f);  // /sqrt(256)

  float mx = scores[0];
  for (int s = 1; s < S_; ++s) mx = fmaxf(mx, scores[s]);
  float pr[S_], esum = 0.f;
  for (int s = 0; s < S_; ++s) { pr[s] = __expf(scores[s] - mx); esum += pr[s]; }
  float pooled = -INFINITY;
  for (int s = 0; s < S_; ++s) { pr[s] /= esum; pooled = fmaxf(pooled, pr[s] * cs[s]); }
  pooled_out[(size_t)b * D_ + d] = pooled;

  if (d == 0) {
    float e = 0.f;
    for (int s = 0; s < S_; ++s) e -= pr[s] * __logf(fmaxf(pr[s], 1e-12f));
    ent_out[b] = e;
  }

  float nrm[S_];
  for (int s = 0; s < S_; ++s)
    nrm[s] = fmaxf(sqrtf(blockReduceSum(cs[s] * cs[s], red)), 1e-12f);
  float o = 0.f;
  for (int s = 0; s < S_; ++s)
    for (int t = s + 1; t < S_; ++t) {
      float sim = blockReduceSum(cs[s] * cs[t], red) / (nrm[s] * nrm[t]);
      o += sim * sim;
    }
  if (d == 0) orth_out[b] = o * (1.f / 10.f);  // mean over S*(S-1)/2 pairs
}

// ============ kernel 2: WMMA GEMM with streaming online logsumexp ============
// grid = (NVS, 64 row-tiles); 128 threads = 4 waves, each wave owns one 16x16
// column tile at a time (V_WMMA_F32_16X16X4_F32, 64 WMMAs per tile over K=256).
// Per-row state (max, sumexp, sumlogits, ctx-select) is carried branchlessly in
// registers, folded over lanes with shfl_xor, waves via LDS, slices in kernel 3.
__global__ __launch_bounds__(128) void gemm_lse_kernel(
    const float* __restrict__ OE, const float* __restrict__ pooled,
    const int* __restrict__ context_ids, float* __restrict__ part) {
  __shared__ float Alds[16][260];                 // +4 pad: stride 1040B (16B aligned, bank-spread)
  __shared__ float Blds[WAVES_PER_BLK][16][260];
  __shared__ float waveRed[WAVES_PER_BLK][16][4];

  int tid = threadIdx.x;
  int wave = tid >> 5, lane = tid & 31;
  int l15 = lane & 15, half = lane >> 4;
  int slice = blockIdx.x;
  int rowbase = blockIdx.y * 16;

  // stage A tile (16 rows of pooled) into LDS, float4 coalesced
  for (int i = 0; i < 8; ++i) {
    int idx = tid + 128 * i;
    int row = idx >> 6, off4 = idx & 63;
    float4 v = *(const float4*)(pooled + (size_t)(rowbase + row) * D_ + off4 * 4);
    *(float4*)(&Alds[row][off4 * 4]) = v;
  }
  __syncthreads();

  int cids[8];
  for (int r = 0; r < 8; ++r) cids[r] = context_ids[rowbase + 8 * half + r];

  float mrun[8], srun[8], trun[8], ctxv[8];
  for (int r = 0; r < 8; ++r) {
    mrun[r] = -INFINITY; srun[r] = 0.f; trun[r] = 0.f; ctxv[r] = 0.f;
  }

  int ctStart = slice * CT_PER_SLICE;
  int ctEnd = ctStart + CT_PER_SLICE;

  for (int it = 0; it < ITERS; ++it) {
    int ct = ctStart + wave + WAVES_PER_BLK * it;
    bool active = ct < ctEnd;
    int colbase = ct * 16;
    __syncthreads();  // uniform: previous reads done before B overwrite
    if (active) {
      // stage 16 OE columns (rows of B^T) into this wave's LDS region
      for (int i = 0; i < 32; ++i) {
        int idx = lane + 32 * i;
        int row = idx >> 6, off4 = idx & 63;
        float4 v = *(const float4*)(OE + (size_t)(colbase + row) * D_ + off4 * 4);
        *(float4*)(&Blds[wave][row][off4 * 4]) = v;
      }
      int nct = ct + WAVES_PER_BLK;
      if (nct < ctEnd)  // hint next tile into cache (global_prefetch_b8)
        __builtin_prefetch(OE + (size_t)(nct * 16 + l15) * D_ + half * 128, 0, 1);
    }
    __syncthreads();  // uniform: B visible before compute
    if (active) {
      v8f c = {0.f, 0.f, 0.f, 0.f, 0.f, 0.f, 0.f, 0.f};
      // f32 A 16x4: lane m=l15 holds K={2*half, 2*half+1}; B 4x16 mirrors with n=l15
      const float* ap = &Alds[l15][2 * half];
      const float* bp = &Blds[wave][l15][2 * half];
#pragma unroll 8
      for (int kk = 0; kk < D_; kk += 4) {
        v2f a;  a[0] = ap[kk];  a[1] = ap[kk + 1];
        v2f bb; bb[0] = bp[kk]; bb[1] = bp[kk + 1];
        c = __builtin_amdgcn_wmma_f32_16x16x4_f32(
            /*neg_a=*/false, a, /*neg_b=*/false, bb,
            /*c_mod=*/(short)0, c, /*reuse_a=*/false, /*reuse_b=*/false);
      }
      // C/D layout: lane holds rows M = 8*half + r, col N = colbase + l15
      int col = colbase + l15;
      for (int r = 0; r < 8; ++r) {
        float val = c[r];
        // branchless online logsumexp update (exp co-executes with WMMA pipe):
        // nm = max(m,val); s = s*exp(m-nm) + exp(val-nm). Handles m=-INF: 0*0+1.
        float nm = fmaxf(mrun[r], val);
        srun[r] = srun[r] * __expf(mrun[r] - nm) + __expf(val - nm);
        mrun[r] = nm;
        trun[r] += val;
        // branchless ctx-logit harvest: exactly one (lane,tile,slice) matches
        ctxv[r] += (cids[r] == col) ? val : 0.f;  // v_cndmask + v_add, no exec churn
      }
    }
  }

  // merge (m, sumexp, sum, ctx) across the 16 lanes that share each C row
  for (int mask = 1; mask <= 8; mask <<= 1) {
    for (int r = 0; r < 8; ++r) {
      float om = __shfl_xor(mrun[r], mask, 32);
      float os = __shfl_xor(srun[r], mask, 32);
      float ot = __shfl_xor(trun[r], mask, 32);
      float oc = __shfl_xor(ctxv[r], mask, 32);
      float M2 = fmaxf(mrun[r], om);
      srun[r] = srun[r] * __expf(mrun[r] - M2) + os * __expf(om - M2);
      mrun[r] = M2;
      trun[r] += ot;
      ctxv[r] += oc;
    }
  }
  if (l15 == 0) {
    for (int r = 0; r < 8; ++r) {
      int row = 8 * half + r;
      waveRed[wave][row][0] = mrun[r];
      waveRed[wave][row][1] = srun[r];
      waveRed[wave][row][2] = trun[r];
      waveRed[wave][row][3] = ctxv[r];
    }
  }
  __syncthreads();
  if (tid < 16) {  // merge the 4 waves, emit per-(slice,row) partial
    float M = -INFINITY, Ssum = 0.f, Tsum = 0.f, Csum = 0.f;
    for (int w = 0; w < WAVES_PER_BLK; ++w) {
      float mw = waveRed[w][tid][0], sw = waveRed[w][tid][1];
      float tw = waveRed[w][tid][2], cw = waveRed[w][tid][3];
      float M2 = fmaxf(M, mw);
      Ssum = Ssum * __expf(M - M2) + sw * __expf(mw - M2);
      M = M2; Tsum += tw; Csum += cw;
    }
    int b = rowbase + tid;
    size_t o = ((size_t)slice * B_ + b) * 4;
    part[o] = M; part[o + 1] = Ssum; part[o + 2] = Tsum; part[o + 3] = Csum;
  }
}

// ============ kernel 3: merge V-slices, per-row w2v loss term ============
__global__ __launch_bounds__(256) void lse_finalize_kernel(
    const float* __restrict__ part, float* __restrict__ w2v) {
  int b = blockIdx.x * 256 + threadIdx.x;
  float M = -INFINITY, Ssum = 0.f, Tsum = 0.f, Csum = 0.f;
  for (int sl = 0; sl < NVS; ++sl) {
    size_t o = ((size_t)sl * B_ + b) * 4;
    float mw = part[o], sw = part[o + 1], tw = part[o + 2], cw = part[o + 3];
    float M2 = fmaxf(M, mw);
    Ssum = Ssum * __expf(M - M2) + sw * __expf(mw - M2);
    M = M2; Tsum += tw; Csum += cw;
  }
  float lse = M + __logf(Ssum);
  float nll = lse - Csum;                            // -logp[ctx]; Csum == logits[b,ctx]
  float neg_mean_logp = lse - Tsum * (1.f / (float)VOCAB);
  w2v[b] = (1.f - LS_) * nll + LS_ * neg_mean_logp;
}

// ============ kernel 4: L2 regularizer (streaming float4 sumsq) ============
__global__ __launch_bounds__(256) void l2_kernel(
    const float* __restrict__ SE, const float* __restrict__ OE,
    float* __restrict__ l2b) {
  __shared__ float red[256];
  const size_t n1 = (size_t)VOCAB * S_ * D_ / 4;  // 16M float4
  const size_t n2 = (size_t)VOCAB * D_ / 4;       // 3.2M float4
  const float4* p1 = (const float4*)SE;
  const float4* p2 = (const float4*)OE;
  float acc = 0.f;
  for (size_t i = (size_t)blockIdx.x * 256 + threadIdx.x; i < n1 + n2;
       i += (size_t)NB2 * 256) {
    float4 v = (i < n1) ? p1[i] : p2[i - n1];
    acc += v.x * v.x + v.y * v.y + v.z * v.z + v.w * v.w;
  }
  float t = blockReduceSum(acc, red);
  if (threadIdx.x == 0) l2b[blockIdx.x] = t;
}

// ============ kernel 5: deterministic final combine ============
__global__ __launch_bounds__(256) void final_kernel(
    const float* __restrict__ w2v, const float* __restrict__ ent,
    const float* __restrict__ orth, const float* __restrict__ l2b,
    float* __restrict__ out) {
  __shared__ float red[256];
  int tid = threadIdx.x;
  float a = 0.f, e = 0.f, o = 0.f, l = 0.f;
  for (int i = tid; i < B_; i += 256) { a += w2v[i]; e += ent[i]; o += orth[i]; }
  for (int i = tid; i < NB2; i += 256) l += l2b[i];
  a = blockReduceSum(a, red);
  e = blockReduceSum(e, red);
  o = blockReduceSum(o, red);
  l = blockReduceSum(l, red);
  if (tid == 0)
    out[0] = A_W2V_ * (a / (float)B_) + A_ORTH_ * (o / (float)B_) +
             A_ENT_ * (e / (float)B_) + A_L2_ * l;
}

extern "C" void kernel_launch(void* const* d_in, const int* in_sizes, int n_in,
                              void* d_out, int out_size, void* d_ws, size_t ws_size,
                              hipStream_t stream) {
  const float* SE = (const float*)d_in[0];          // [VOCAB,S,D] f32
  const float* OE = (const float*)d_in[1];          // [VOCAB,D]   f32
  const int* center_pos = (const int*)d_in[2];      // [B]
  const int* context_ids = (const int*)d_in[3];     // [B]
  const int* qids = (const int*)d_in[4];            // [B,T]

  float* ws = (float*)d_ws;
  float* pooled = ws + OFF_POOLED;
  float* w2v    = ws + OFF_W2V;
  float* ent    = ws + OFF_ENT;
  float* orth   = ws + OFF_ORTH;
  float* l2b    = ws + OFF_L2B;
  float* part   = ws + OFF_PART;

  gating_kernel<<<B_, 256, 0, stream>>>(SE, center_pos, qids, pooled, ent, orth);
  gemm_lse_kernel<<<dim3(NVS, B_ / 16), 128, 0, stream>>>(OE, pooled, context_ids, part);
  lse_finalize_kernel<<<B_ / 256, 256, 0, stream>>>(part, w2v);
  l2_kernel<<<NB2, 256, 0, stream>>>(SE, OE, l2b);
  final_kernel<<<1, 256, 0, stream>>>(w2v, ent, orth, l2b, (float*)d_out);
}